// deMambaBlock_49838800503535
// MI455X (gfx1250) — compile-verified
//
#include <hip/hip_runtime.h>
#include <hip/hip_bf16.h>
#include <math.h>

typedef __attribute__((ext_vector_type(2))) float v2f;
typedef __attribute__((ext_vector_type(8))) float v8f;

#define D_MODEL  256
#define D_INNER  512
#define D_STATE  64
#define DT_RANK  4
#define D_CONV   4
#define BB       8
#define LL       144
#define M_ROWS   (BB * LL)            // 1152 (divisible by 32)
#define NX       (2 * D_INNER)        // 1024
#define NDBC     (DT_RANK + 2 * D_STATE)  // 132
#define NDBC_PAD 144                  // padded to 9 tiles of 16

__device__ __forceinline__ float siluf(float v)     { return v / (1.0f + __expf(-v)); }
__device__ __forceinline__ float softplusf(float v) { return (v > 20.0f) ? v : log1pf(__expf(v)); }

// ---------------------------------------------------------------------------
// WMMA helpers for Out[m][n] = sum_k X[m][k] * W[n][k]  (X: MxK, W: NxK, both
// row major; W is the transposed B operand).  V_WMMA_F32_16X16X4_F32 layout:
// lane = 16*h + lm holds A[m0+lm][2h..2h+1] / B[2h..2h+1][n0+lm]; C/D vgpr i
// holds row m0 + i + 8*h, col n0 + lm.
// ---------------------------------------------------------------------------
__device__ __forceinline__ v8f wmma4(v2f a, v2f b, v8f c)
{
    return __builtin_amdgcn_wmma_f32_16x16x4_f32(false, a, false, b,
                                                 (short)0, c, false, false);
}

// 32x32 register-blocked tile: 4 WMMA per 4 loads (1 load/WMMA)
__device__ __forceinline__ void wmma_tile_32x32(const float* __restrict__ X,
                                                const float* __restrict__ W,
                                                int K, int m0, int n0,
                                                v8f& c00, v8f& c01,
                                                v8f& c10, v8f& c11)
{
    const int lane = threadIdx.x & 31;
    const int half = lane >> 4;
    const int lm   = lane & 15;

    const float* xrow0 = X + (size_t)(m0 + lm) * K + 2 * half;
    const float* xrow1 = xrow0 + (size_t)16 * K;
    const float* wrow0 = W + (size_t)(n0 + lm) * K + 2 * half;
    const float* wrow1 = wrow0 + (size_t)16 * K;

    c00 = {}; c01 = {}; c10 = {}; c11 = {};
#pragma unroll 4
    for (int k = 0; k < K; k += 4) {
        v2f a0 = *(const v2f*)(xrow0 + k);
        v2f a1 = *(const v2f*)(xrow1 + k);
        v2f b0 = *(const v2f*)(wrow0 + k);
        v2f b1 = *(const v2f*)(wrow1 + k);
        c00 = wmma4(a0, b0, c00);
        c01 = wmma4(a0, b1, c01);
        c10 = wmma4(a1, b0, c10);
        c11 = wmma4(a1, b1, c11);
    }
}

// 16x16 tile with N guard (for the N=132 GEMM); mask by multiply so EXEC
// stays all-1s around the WMMA.
__device__ __forceinline__ v8f wmma_tile_16x16_guard(const float* __restrict__ X,
                                                     const float* __restrict__ W,
                                                     int K, int Nw, int m0, int n0)
{
    const int lane = threadIdx.x & 31;
    const int half = lane >> 4;
    const int lm   = lane & 15;

    const int   n     = n0 + lm;
    const float bmask = (n < Nw) ? 1.0f : 0.0f;
    const int   nsafe = (n < Nw) ? n : 0;

    const float* xrow = X + (size_t)(m0 + lm) * K + 2 * half;
    const float* wrow = W + (size_t)nsafe     * K + 2 * half;

    v8f acc = {};
#pragma unroll 8
    for (int k = 0; k < K; k += 4) {
        v2f a = *(const v2f*)(xrow + k);
        v2f b = *(const v2f*)(wrow + k);
        b.x *= bmask; b.y *= bmask;
        acc = wmma4(a, b, acc);
    }
    return acc;
}

__device__ __forceinline__ void store_tile(float* __restrict__ Out, int ldo,
                                           int m0, int n0, v8f acc)
{
    const int lane = threadIdx.x & 31;
    const int half = lane >> 4;
    const int lm   = lane & 15;
#pragma unroll
    for (int i = 0; i < 8; ++i)
        Out[(size_t)(m0 + i + 8 * half) * ldo + (n0 + lm)] = acc[i];
}

// --- GEMM1: xz = x @ W_in^T, split into ix_raw (cols 0..511) and z ----------
__global__ void gemm_in_kernel(const float* __restrict__ x,
                               const float* __restrict__ W_in,
                               float* __restrict__ ixraw, float* __restrict__ zbuf)
{
    const int NT   = NX / 32;                                    // 32
    const int tile = blockIdx.x * (blockDim.x >> 5) + (threadIdx.x >> 5);
    const int mt   = tile / NT, nt = tile % NT;
    if (mt >= M_ROWS / 32) return;

    v8f c00, c01, c10, c11;
    wmma_tile_32x32(x, W_in, D_MODEL, mt * 32, nt * 32, c00, c01, c10, c11);

    const int n0 = nt * 32;                 // 512 % 32 == 0: tile stays in one buf
    float* dst; int nc;
    if (n0 < D_INNER) { dst = ixraw; nc = n0; }
    else              { dst = zbuf;  nc = n0 - D_INNER; }
    store_tile(dst, D_INNER, mt * 32,      nc,      c00);
    store_tile(dst, D_INNER, mt * 32,      nc + 16, c01);
    store_tile(dst, D_INNER, mt * 32 + 16, nc,      c10);
    store_tile(dst, D_INNER, mt * 32 + 16, nc + 16, c11);
}

// --- depthwise conv (pad left 2, right 1) + SiLU ----------------------------
__global__ void conv_silu_kernel(const float* __restrict__ ixraw,
                                 const float* __restrict__ cw,
                                 const float* __restrict__ cb,
                                 float* __restrict__ ix)
{
    const int idx = blockIdx.x * blockDim.x + threadIdx.x;
    if (idx >= M_ROWS * D_INNER) return;
    const int d = idx % D_INNER;
    const int m = idx / D_INNER;
    const int t = m % LL, b = m / LL;

    float acc = cb[d];
#pragma unroll
    for (int k = 0; k < D_CONV; ++k) {
        const int ti = t + k - 2;
        if (ti >= 0 && ti < LL)
            acc += ixraw[(size_t)(b * LL + ti) * D_INNER + d] * cw[d * D_CONV + k];
    }
    ix[idx] = siluf(acc);
}

// --- GEMM2: dBC = ix @ W_x^T  (N=132, padded to 144 with zero mask) ---------
__global__ void gemm_wx_kernel(const float* __restrict__ ix,
                               const float* __restrict__ W_x,
                               float* __restrict__ dbc)
{
    const int NT   = NDBC_PAD / 16;                              // 9
    const int tile = blockIdx.x * (blockDim.x >> 5) + (threadIdx.x >> 5);
    const int mt   = tile / NT, nt = tile % NT;
    if (mt >= M_ROWS / 16) return;

    v8f acc = wmma_tile_16x16_guard(ix, W_x, D_INNER, NDBC, mt * 16, nt * 16);
    store_tile(dbc, NDBC_PAD, mt * 16, nt * 16, acc);
}

// --- delta = softplus(dt_r @ W_dt^T + b_dt)  (K = 4) ------------------------
__global__ void delta_kernel(const float* __restrict__ dbc,
                             const float* __restrict__ W_dt,
                             const float* __restrict__ b_dt,
                             float* __restrict__ delta)
{
    const int idx = blockIdx.x * blockDim.x + threadIdx.x;
    if (idx >= M_ROWS * D_INNER) return;
    const int d = idx % D_INNER;
    const int m = idx / D_INNER;
    float s = b_dt[d];
#pragma unroll
    for (int r = 0; r < DT_RANK; ++r)
        s += dbc[(size_t)m * NDBC_PAD + r] * W_dt[d * DT_RANK + r];
    delta[idx] = softplusf(s);
}

// --- selective scan: one wave per (b,d); each lane owns states n, n+32 ------
__global__ void scan_kernel(const float* __restrict__ delta,
                            const float* __restrict__ ix,
                            const float* __restrict__ zbuf,
                            const float* __restrict__ dbc,
                            const float* __restrict__ A_log,
                            const float* __restrict__ Dp,
                            float* __restrict__ g)
{
    const int wid  = blockIdx.x * (blockDim.x >> 5) + (threadIdx.x >> 5);
    const int lane = threadIdx.x & 31;
    const int b = wid / D_INNER;
    const int d = wid % D_INNER;
    if (b >= BB) return;

    const float A0 = -__expf(A_log[d * D_STATE + lane]);
    const float A1 = -__expf(A_log[d * D_STATE + lane + 32]);
    const float Dd = Dp[d];
    float h0 = 0.0f, h1 = 0.0f;

    for (int t = 0; t < LL; ++t) {
        const size_t m   = (size_t)b * LL + t;
        const float  dt  = delta[m * D_INNER + d];
        const float  xt  = ix[m * D_INNER + d];
        const float* row = dbc + m * NDBC_PAD;
        const float  Bn0 = row[DT_RANK + lane];
        const float  Bn1 = row[DT_RANK + lane + 32];
        const float  Cn0 = row[DT_RANK + D_STATE + lane];
        const float  Cn1 = row[DT_RANK + D_STATE + lane + 32];

        const float du = dt * xt;
        h0 = __expf(dt * A0) * h0 + du * Bn0;
        h1 = __expf(dt * A1) * h1 + du * Bn1;

        float p = h0 * Cn0 + h1 * Cn1;
#pragma unroll
        for (int off = 16; off > 0; off >>= 1)
            p += __shfl_xor(p, off, 32);

        if (lane == 0) {
            const float zt = zbuf[m * D_INNER + d];
            g[m * D_INNER + d] = (p + Dd * xt) * siluf(zt);
        }
    }
}

// --- GEMM3: out = g @ W_out^T -----------------------------------------------
__global__ void gemm_out_kernel(const float* __restrict__ g,
                                const float* __restrict__ W_out,
                                float* __restrict__ out)
{
    const int NT   = D_MODEL / 32;                               // 8
    const int tile = blockIdx.x * (blockDim.x >> 5) + (threadIdx.x >> 5);
    const int mt   = tile / NT, nt = tile % NT;
    if (mt >= M_ROWS / 32) return;

    v8f c00, c01, c10, c11;
    wmma_tile_32x32(g, W_out, D_INNER, mt * 32, nt * 32, c00, c01, c10, c11);

    store_tile(out, D_MODEL, mt * 32,      nt * 32,      c00);
    store_tile(out, D_MODEL, mt * 32,      nt * 32 + 16, c01);
    store_tile(out, D_MODEL, mt * 32 + 16, nt * 32,      c10);
    store_tile(out, D_MODEL, mt * 32 + 16, nt * 32 + 16, c11);
}

extern "C" void kernel_launch(void* const* d_in, const int* in_sizes, int n_in,
                              void* d_out, int out_size, void* d_ws, size_t ws_size,
                              hipStream_t stream)
{
    const float* x      = (const float*)d_in[0];
    // d_in[1] = lastin: unused (reference starts the scan from h0 = zeros)
    const float* W_in   = (const float*)d_in[2];
    const float* conv_w = (const float*)d_in[3];
    const float* conv_b = (const float*)d_in[4];
    const float* W_x    = (const float*)d_in[5];
    const float* W_dt   = (const float*)d_in[6];
    const float* b_dt   = (const float*)d_in[7];
    const float* A_log  = (const float*)d_in[8];
    const float* Dp     = (const float*)d_in[9];
    const float* W_out  = (const float*)d_in[10];
    float* out = (float*)d_out;

    float* ws = (float*)d_ws;
    const size_t NE = (size_t)M_ROWS * D_INNER;   // 589824 floats
    float* ixraw = ws; ws += NE;
    float* zbuf  = ws; ws += NE;
    float* ixc   = ws; ws += NE;
    float* dbc   = ws; ws += (size_t)M_ROWS * NDBC_PAD;
    float* dl    = ws; ws += NE;
    float* g     = ws; ws += NE;

    {   // GEMM1: (1152/32) x (1024/32) = 36 x 32 tiles, 4 waves/block
        const int tiles = (M_ROWS / 32) * (NX / 32);
        gemm_in_kernel<<<tiles / 4, 128, 0, stream>>>(x, W_in, ixraw, zbuf);
    }
    conv_silu_kernel<<<(int)((NE + 255) / 256), 256, 0, stream>>>(ixraw, conv_w, conv_b, ixc);
    {   // GEMM2: 72 x 9 tiles of 16x16
        const int tiles = (M_ROWS / 16) * (NDBC_PAD / 16);
        gemm_wx_kernel<<<(tiles + 3) / 4, 128, 0, stream>>>(ixc, W_x, dbc);
    }
    delta_kernel<<<(int)((NE + 255) / 256), 256, 0, stream>>>(dbc, W_dt, b_dt, dl);
    {   // scan: one wave per (b,d) = 4096 waves
        const int waves = BB * D_INNER;
        scan_kernel<<<waves / 4, 128, 0, stream>>>(dl, ixc, zbuf, dbc, A_log, Dp, g);
    }
    {   // GEMM3: 36 x 8 tiles of 32x32
        const int tiles = (M_ROWS / 32) * (D_MODEL / 32);
        gemm_out_kernel<<<tiles / 4, 128, 0, stream>>>(g, W_out, out);
    }
}